// Captioner_41412074668572
// MI455X (gfx1250) — compile-verified
//
#include <hip/hip_runtime.h>

// Problem constants (from reference): B=32, N=2048, L=128, D=1024, V=32000
#define B_  32
#define N_  2048
#define L_  128
#define D_  1024
#define KC  32      // K-chunk staged in LDS per iteration (double-buffered)
#define LDP 36      // padded LDS row stride (floats): 144B, 16B-aligned, n*36%64 distinct
#define NCHUNK (D_ / KC)

typedef __attribute__((ext_vector_type(2))) float v2f;
typedef __attribute__((ext_vector_type(8))) float v8f;

#if __has_builtin(__builtin_amdgcn_global_load_async_to_lds_b128) && \
    __has_builtin(__builtin_amdgcn_global_load_async_to_lds_b64)  && \
    __has_builtin(__builtin_amdgcn_s_wait_asynccnt)
#define USE_ASYNC_LDS 1
typedef int v2i_vs __attribute__((vector_size(8)));    // b64 param pointee type
typedef int v4i_vs __attribute__((vector_size(16)));   // b128 param pointee type
typedef __attribute__((address_space(3))) v2i_vs lds_v2i;   // __shared__ v2i
typedef __attribute__((address_space(3))) v4i_vs lds_v4i;   // __shared__ v4i
#else
#define USE_ASYNC_LDS 0
#endif

__global__ void zero_out_kernel(float* out) {
    if (threadIdx.x == 0) out[0] = 0.0f;
}

// x2[row] = sum_d img[row,d]^2 ; one wave32 per row, 8 rows per block
__global__ __launch_bounds__(256) void rowsq_img_kernel(const float* __restrict__ img,
                                                        float* __restrict__ x2) {
    const int wave = threadIdx.x >> 5, lane = threadIdx.x & 31;
    const int row  = blockIdx.x * 8 + wave;              // < B*N
    const float4* p = (const float4*)(img + (size_t)row * D_);
    float s = 0.0f;
    #pragma unroll
    for (int i = 0; i < D_ / 4 / 32; ++i) {              // 8 x float4 per lane
        float4 v = p[lane + 32 * i];
        s += v.x * v.x + v.y * v.y + v.z * v.z + v.w * v.w;
    }
    #pragma unroll
    for (int m = 16; m >= 1; m >>= 1) s += __shfl_xor(s, m, 32);
    if (lane == 0) x2[row] = s;
}

// y2[row] = sum_d emb[tok[row],d]^2 ; one wave32 per (b,l) row
__global__ __launch_bounds__(256) void rowsq_emb_kernel(const float* __restrict__ emb,
                                                        const int* __restrict__ tok,
                                                        float* __restrict__ y2) {
    const int wave = threadIdx.x >> 5, lane = threadIdx.x & 31;
    const int row  = blockIdx.x * 8 + wave;              // < B*L
    const int tid  = tok[row];
    const float4* p = (const float4*)(emb + (size_t)tid * D_);
    float s = 0.0f;
    #pragma unroll
    for (int i = 0; i < D_ / 4 / 32; ++i) {
        float4 v = p[lane + 32 * i];
        s += v.x * v.x + v.y * v.y + v.z * v.z + v.w * v.w;
    }
    #pragma unroll
    for (int m = 16; m >= 1; m >>= 1) s += __shfl_xor(s, m, 32);
    if (lane == 0) y2[row] = s;
}

// Main fused kernel: per block -> 16 rows x 128 cols of the [N x L] distance
// tile for one batch. 8 waves, wave w owns column tile [16w, 16w+16).
// GEMM via v_wmma_f32_16x16x4_f32; K staged through double-buffered LDS using
// CDNA5 async global->LDS loads (ASYNCcnt) when available.
__global__ __launch_bounds__(256) void cdist_min_kernel(
    const float* __restrict__ img, const int* __restrict__ tok,
    const float* __restrict__ emb, const float* __restrict__ x2,
    const float* __restrict__ y2,  float* __restrict__ out) {

    __shared__ float sA[2][16 * LDP];    // 16 rows   x KC, double buffered
    __shared__ float sB[2][L_ * LDP];    // 128 cols  x KC, double buffered (gathered)
    __shared__ int   stid[L_];
    __shared__ float sx2[16];
    __shared__ float sy2[L_];
    __shared__ float wmin[8][16];
    __shared__ float rowmin[16];

    const int t    = threadIdx.x;
    const int lane = t & 31;
    const int wave = t >> 5;
    const int b    = blockIdx.y;
    const int row0 = blockIdx.x * 16;

    if (t < L_) {
        stid[t] = tok[b * L_ + t];
        sy2[t]  = y2[b * L_ + t];
    }
    if (t < 16) sx2[t] = x2[b * N_ + row0 + t];
    __syncthreads();

    v8f acc = {};                               // 16x16 f32 accumulator (8 VGPRs)
    const int col0 = wave * 16;
    const int n    = lane & 15;                 // N index within tile (lane)
    const int kh   = (lane >> 4) * 2;           // K sub-offset per 16-lane half
    const int m    = n;                         // A-matrix M index = lane % 16

    const float* Abase = img + ((size_t)b * N_ + row0) * D_;

    // Per-chunk staging addresses (uniform per thread across chunks).
    const int ar  = t >> 4;                     // A: row 0..15
    const int ac  = (t & 15) * 2;               // A: col pair 0..30
    // B: 4 float4 per thread: idx = t + 256*i, row = idx/8, c4 = idx%8

#if USE_ASYNC_LDS
    // Issue the 5 async global->LDS loads for chunk (k0) into buffer buf.
    auto issue_chunk = [&](int k0, int buf) {
        __builtin_amdgcn_global_load_async_to_lds_b64(
            (v2i_vs*)(Abase + (size_t)ar * D_ + k0 + ac),
            (lds_v2i*)&sA[buf][ar * LDP + ac], 0, 0);
        #pragma unroll
        for (int i = 0; i < 4; ++i) {
            const int idx = t + 256 * i;
            const int row = idx >> 3, c4 = idx & 7;
            __builtin_amdgcn_global_load_async_to_lds_b128(
                (v4i_vs*)(emb + (size_t)stid[row] * D_ + k0 + c4 * 4),
                (lds_v4i*)&sB[buf][row * LDP + c4 * 4], 0, 0);
        }
    };

    issue_chunk(0, 0);
    for (int k = 0; k < NCHUNK; ++k) {
        const int cur = k & 1;
        if (k + 1 < NCHUNK) {
            issue_chunk((k + 1) * KC, cur ^ 1);
            __builtin_amdgcn_s_wait_asynccnt(5);   // 5 newest (next chunk) may remain
        } else {
            __builtin_amdgcn_s_wait_asynccnt(0);
        }
        __syncthreads();
        #pragma unroll
        for (int kk = 0; kk < KC; kk += 4) {
            v2f a, bb;
            a.x  = sA[cur][m * LDP + kk + kh];
            a.y  = sA[cur][m * LDP + kk + kh + 1];
            bb.x = sB[cur][(col0 + n) * LDP + kk + kh];
            bb.y = sB[cur][(col0 + n) * LDP + kk + kh + 1];
            acc = __builtin_amdgcn_wmma_f32_16x16x4_f32(
                false, a, false, bb, (short)0, acc, false, false);
        }
        __syncthreads();
    }
#else
    // Fallback: register-batched staging (loads issued before the barrier so
    // latency overlaps the previous iteration's tail).
    for (int k = 0; k < NCHUNK; ++k) {
        const int k0 = k * KC;
        const int cur = k & 1;
        float2 va = *(const float2*)(Abase + (size_t)ar * D_ + k0 + ac);
        float4 vb[4];
        #pragma unroll
        for (int i = 0; i < 4; ++i) {
            const int idx = t + 256 * i;
            const int row = idx >> 3, c4 = idx & 7;
            vb[i] = *(const float4*)(emb + (size_t)stid[row] * D_ + k0 + c4 * 4);
        }
        if (k + 1 < NCHUNK)
            __builtin_prefetch(Abase + (size_t)ar * D_ + k0 + KC, 0, 3);
        *(float2*)(&sA[cur][ar * LDP + ac]) = va;
        #pragma unroll
        for (int i = 0; i < 4; ++i) {
            const int idx = t + 256 * i;
            const int row = idx >> 3, c4 = idx & 7;
            *(float4*)(&sB[cur][row * LDP + c4 * 4]) = vb[i];
        }
        __syncthreads();
        #pragma unroll
        for (int kk = 0; kk < KC; kk += 4) {
            v2f a, bb;
            a.x  = sA[cur][m * LDP + kk + kh];
            a.y  = sA[cur][m * LDP + kk + kh + 1];
            bb.x = sB[cur][(col0 + n) * LDP + kk + kh];
            bb.y = sB[cur][(col0 + n) * LDP + kk + kh + 1];
            acc = __builtin_amdgcn_wmma_f32_16x16x4_f32(
                false, a, false, bb, (short)0, acc, false, false);
        }
        __syncthreads();
    }
#endif

    // Epilogue: cost = sqrt(max(x2 + y2 - 2*xy, 0)); min over the L dimension.
    // C layout (32-bit 16x16): lanes 0-15 -> N=lane, VGPR r -> M=r;
    //                          lanes 16-31 -> N=lane-16, VGPR r -> M=r+8.
    const float y2v = sy2[col0 + n];
    float mn[8];
    #pragma unroll
    for (int r = 0; r < 8; ++r) {
        const int M = r + ((lane >> 4) << 3);
        const float d2 = sx2[M] + y2v - 2.0f * acc[r];
        mn[r] = sqrtf(fmaxf(d2, 0.0f));
    }
    #pragma unroll
    for (int r = 0; r < 8; ++r) {
        float v = mn[r];
        #pragma unroll
        for (int msk = 8; msk >= 1; msk >>= 1)
            v = fminf(v, __shfl_xor(v, msk, 32));
        mn[r] = v;
    }
    if (lane == 0 || lane == 16) {
        const int Mb = (lane >> 4) << 3;
        #pragma unroll
        for (int r = 0; r < 8; ++r) wmin[wave][Mb + r] = mn[r];
    }
    __syncthreads();

    if (t < 16) {
        float v = wmin[0][t];
        #pragma unroll
        for (int w = 1; w < 8; ++w) v = fminf(v, wmin[w][t]);
        rowmin[t] = v;
    }
    __syncthreads();

    if (t == 0) {
        float s = 0.0f;
        #pragma unroll
        for (int i = 0; i < 16; ++i) s += rowmin[i];
        atomicAdd(out, s * (1.0f / ((float)B_ * (float)N_)));
    }
}

extern "C" void kernel_launch(void* const* d_in, const int* in_sizes, int n_in,
                              void* d_out, int out_size, void* d_ws, size_t ws_size,
                              hipStream_t stream) {
    const float* img = (const float*)d_in[0];   // image_features [B,N,D] f32
    const int*   tok = (const int*)d_in[1];     // token_ids      [B,L]
    const float* emb = (const float*)d_in[2];   // emb_table      [V,D] f32
    float* out = (float*)d_out;                 // scalar mean

    float* x2 = (float*)d_ws;                   // [B*N] floats (256 KB)
    float* y2 = x2 + (size_t)B_ * N_;           // [B*L] floats (16 KB)

    zero_out_kernel<<<1, 64, 0, stream>>>(out);
    rowsq_img_kernel<<<(B_ * N_) / 8, 256, 0, stream>>>(img, x2);
    rowsq_emb_kernel<<<(B_ * L_) / 8, 256, 0, stream>>>(emb, tok, y2);

    dim3 grid(N_ / 16, B_);
    cdist_min_kernel<<<grid, 256, 0, stream>>>(img, tok, emb, x2, y2, out);
}